// GATEncoder_82892868813180
// MI455X (gfx1250) — compile-verified
//
#include <hip/hip_runtime.h>
#include <math.h>

typedef __attribute__((ext_vector_type(2))) float v2f;
typedef __attribute__((ext_vector_type(8))) float v8f;

#define D 64
#define NEG_SLOPE 0.2f

// ---------------------------------------------------------------------------
// Order-preserving float<->uint encoding for atomicMax on floats.
// enc is monotonic: a < b  <=>  enc(a) < enc(b) (unsigned). 0u < enc(any finite).
__device__ __forceinline__ unsigned enc_f32(float f) {
  unsigned u = __float_as_uint(f);
  return (u & 0x80000000u) ? ~u : (u | 0x80000000u);
}
__device__ __forceinline__ float dec_f32(unsigned u) {
  return __uint_as_float((u & 0x80000000u) ? (u & 0x7fffffffu) : ~u);
}

// ---------------------------------------------------------------------------
// xl = h @ Wl + bl ; xr = h @ Wr + br  via V_WMMA_F32_16X16X4_F32.
// One wave computes a 16-row x 64-col strip of one output (blockIdx.y selects l/r).
// A-frag (16x4 f32): lanes 0-15 -> M=lane, VGPR0=K0,VGPR1=K1; lanes 16-31 -> K2,K3.
// B-frag (4x16 f32): lanes 0-15 -> N=lane, VGPR0=rowK0,VGPR1=rowK1; lanes 16-31 -> rowK2,rowK3.
// D-frag (16x16 f32): VGPR i: lanes0-15 M=i, lanes16-31 M=i+8; N = lane&15.
__global__ void gemm_xlxr_wmma(const float* __restrict__ h,
                               const float* __restrict__ Wl, const float* __restrict__ bl,
                               const float* __restrict__ Wr, const float* __restrict__ br,
                               float* __restrict__ xl, float* __restrict__ xr,
                               int n_rows) {
  const float* W    = (blockIdx.y == 0) ? Wl : Wr;
  const float* bias = (blockIdx.y == 0) ? bl : br;
  float*       out  = (blockIdx.y == 0) ? xl : xr;

  int wave = blockIdx.x * (blockDim.x >> 5) + (threadIdx.x >> 5);
  int ntiles = n_rows >> 4;            // full 16-row tiles
  if (wave >= ntiles) return;          // wave-granular: EXEC stays all-1s for WMMA

  int r0   = wave << 4;
  int lane = threadIdx.x & 31;
  int half = lane >> 4;                // 0: lanes 0-15, 1: lanes 16-31
  int l15  = lane & 15;

  v8f acc0 = {}, acc1 = {}, acc2 = {}, acc3 = {};
  const float* arow = h + (size_t)(r0 + l15) * D;

  for (int k0 = 0; k0 < D; k0 += 4) {
    int ka = k0 + (half << 1);         // this lane-half's K base
    v2f A;
    A.x = arow[ka];
    A.y = arow[ka + 1];
    const float* w0 = W + (size_t)ka * D;        // weight row K=ka
    const float* w1 = W + (size_t)(ka + 1) * D;  // weight row K=ka+1
    v2f B0, B1, B2, B3;
    B0.x = w0[l15];      B0.y = w1[l15];
    B1.x = w0[16 + l15]; B1.y = w1[16 + l15];
    B2.x = w0[32 + l15]; B2.y = w1[32 + l15];
    B3.x = w0[48 + l15]; B3.y = w1[48 + l15];
    acc0 = __builtin_amdgcn_wmma_f32_16x16x4_f32(false, A, false, B0, (short)0, acc0, false, false);
    acc1 = __builtin_amdgcn_wmma_f32_16x16x4_f32(false, A, false, B1, (short)0, acc1, false, false);
    acc2 = __builtin_amdgcn_wmma_f32_16x16x4_f32(false, A, false, B2, (short)0, acc2, false, false);
    acc3 = __builtin_amdgcn_wmma_f32_16x16x4_f32(false, A, false, B3, (short)0, acc3, false, false);
  }

#pragma unroll
  for (int i = 0; i < 8; ++i) {
    int row = r0 + i + (half << 3);
    float* orow = out + (size_t)row * D;
    orow[l15]      = acc0[i] + bias[l15];
    orow[16 + l15] = acc1[i] + bias[16 + l15];
    orow[32 + l15] = acc2[i] + bias[32 + l15];
    orow[48 + l15] = acc3[i] + bias[48 + l15];
  }
}

// Scalar tail for n_rows % 16 (not hit for N=50000 but keeps kernel general).
__global__ void gemm_tail(const float* __restrict__ h,
                          const float* __restrict__ Wl, const float* __restrict__ bl,
                          const float* __restrict__ Wr, const float* __restrict__ br,
                          float* __restrict__ xl, float* __restrict__ xr,
                          int row_start, int n_rows) {
  const float* W    = (blockIdx.y == 0) ? Wl : Wr;
  const float* bias = (blockIdx.y == 0) ? bl : br;
  float*       out  = (blockIdx.y == 0) ? xl : xr;
  int idx = blockIdx.x * blockDim.x + threadIdx.x;   // (row-row_start)*D + col
  int row = row_start + idx / D;
  int col = idx % D;
  if (row >= n_rows) return;
  float s = bias[col];
  const float* hr = h + (size_t)row * D;
  for (int k = 0; k < D; ++k) s += hr[k] * W[(size_t)k * D + col];
  out[(size_t)row * D + col] = s;
}

// ---------------------------------------------------------------------------
__global__ void init_layer(float* __restrict__ acc, unsigned* __restrict__ nmax,
                           float* __restrict__ denom, int n) {
  int idx = blockIdx.x * blockDim.x + threadIdx.x;
  if (idx >= n * D) return;
  acc[idx] = 0.0f;
  if ((idx & (D - 1)) == 0) {
    nmax[idx >> 6]  = 0u;     // below enc() of any finite float
    denom[idx >> 6] = 0.0f;
  }
}

// Pass A: per-edge GATv2 score e = leaky_relu(xl[src]+xr[dst]) . att ; seg-max.
// One wave per edge; lane owns features 2*lane, 2*lane+1.
__global__ void edge_score(const float* __restrict__ xl, const float* __restrict__ xr,
                           const float* __restrict__ att, const int* __restrict__ ei,
                           int E, int EP, float* __restrict__ e_out,
                           unsigned* __restrict__ nmax) {
  int wave = blockIdx.x * (blockDim.x >> 5) + (threadIdx.x >> 5);
  if (wave >= EP) return;
  int lane = threadIdx.x & 31;
  int src, dst;
  if (wave < E) { src = ei[wave]; dst = ei[E + wave]; }
  else          { src = dst = wave - E; }           // self-loop
  int c = lane << 1;
  float s0 = xl[(size_t)src * D + c]     + xr[(size_t)dst * D + c];
  float s1 = xl[(size_t)src * D + c + 1] + xr[(size_t)dst * D + c + 1];
  s0 = (s0 > 0.0f) ? s0 : NEG_SLOPE * s0;
  s1 = (s1 > 0.0f) ? s1 : NEG_SLOPE * s1;
  float p = s0 * att[c] + s1 * att[c + 1];
#pragma unroll
  for (int off = 16; off > 0; off >>= 1)
    p += __shfl_xor(p, off, 32);
  if (lane == 0) {
    e_out[wave] = p;
    atomicMax(&nmax[dst], enc_f32(p));
  }
}

// Pass B: ex = exp(e - m[dst]); denom[dst] += ex; acc[dst] += ex * xl[src].
__global__ void edge_accum(const float* __restrict__ xl, const int* __restrict__ ei,
                           int E, int EP, const float* __restrict__ e_in,
                           const unsigned* __restrict__ nmax,
                           float* __restrict__ denom, float* __restrict__ acc) {
  int wave = blockIdx.x * (blockDim.x >> 5) + (threadIdx.x >> 5);
  if (wave >= EP) return;
  int lane = threadIdx.x & 31;
  int src, dst;
  if (wave < E) { src = ei[wave]; dst = ei[E + wave]; }
  else          { src = dst = wave - E; }
  float m  = dec_f32(nmax[dst]);
  float ex = __expf(e_in[wave] - m);
  if (lane == 0) atomicAdd(&denom[dst], ex);
  int c = lane << 1;
  atomicAdd(&acc[(size_t)dst * D + c],     ex * xl[(size_t)src * D + c]);
  atomicAdd(&acc[(size_t)dst * D + c + 1], ex * xl[(size_t)src * D + c + 1]);
}

// out = relu(acc/denom + bias) + h_in  (relu + residual)
__global__ void finalize(const float* __restrict__ acc, const float* __restrict__ denom,
                         const float* __restrict__ bias, const float* __restrict__ h_in,
                         float* __restrict__ h_out, int n) {
  int idx = blockIdx.x * blockDim.x + threadIdx.x;
  if (idx >= n * D) return;
  int node = idx >> 6;
  int c    = idx & (D - 1);
  float v = acc[idx] / denom[node] + bias[c];
  v = (v > 0.0f) ? v : 0.0f;
  h_out[idx] = v + h_in[idx];
}

// ---------------------------------------------------------------------------
extern "C" void kernel_launch(void* const* d_in, const int* in_sizes, int n_in,
                              void* d_out, int out_size, void* d_ws, size_t ws_size,
                              hipStream_t stream) {
  const float* x    = (const float*)d_in[0];
  const int*   ei   = (const int*)d_in[1];   // [2, E] (src row then dst row)
  const float* Wl   = (const float*)d_in[2];
  const float* bl   = (const float*)d_in[3];
  const float* Wr   = (const float*)d_in[4];
  const float* br   = (const float*)d_in[5];
  const float* att  = (const float*)d_in[6];
  const float* bias = (const float*)d_in[7];

  const int n  = in_sizes[0] / D;
  const int E  = in_sizes[1] / 2;
  const int L  = in_sizes[2] / (D * D);
  const int EP = E + n;                       // edges incl. self-loops
  const size_t nd = (size_t)n * D;

  float*    ws    = (float*)d_ws;
  float*    xl    = ws;                       // [n, D]
  float*    xr    = xl + nd;                  // [n, D]
  float*    acc   = xr + nd;                  // [n, D]
  float*    e_buf = acc + nd;                 // [EP]
  unsigned* nmax  = (unsigned*)(e_buf + EP);  // [n]
  float*    denom = (float*)(nmax + n);       // [n]
  float*    hA    = denom + n;                // [n, D]
  float*    hB    = hA + nd;                  // [n, D]

  const int row_tiles = n >> 4;
  const int row_rem   = n & 15;
  const int tot       = n * D;

  const float* h_in = x;
  for (int l = 0; l < L; ++l) {
    float* h_out = (l == L - 1) ? (float*)d_out : ((l & 1) ? hB : hA);
    const float* Wl_l = Wl + (size_t)l * D * D;
    const float* Wr_l = Wr + (size_t)l * D * D;

    // 1) xl/xr transforms (WMMA fp32)
    if (row_tiles > 0) {
      dim3 grid((row_tiles + 7) / 8, 2, 1);
      gemm_xlxr_wmma<<<grid, 256, 0, stream>>>(h_in, Wl_l, bl + l * D, Wr_l, br + l * D,
                                               xl, xr, n);
    }
    if (row_rem > 0) {
      dim3 grid((row_rem * D + 255) / 256, 2, 1);
      gemm_tail<<<grid, 256, 0, stream>>>(h_in, Wl_l, bl + l * D, Wr_l, br + l * D,
                                          xl, xr, row_tiles << 4, n);
    }
    // 2) zero accumulators / max / denom
    init_layer<<<(tot + 255) / 256, 256, 0, stream>>>(acc, nmax, denom, n);
    // 3) edge scores + segment max
    edge_score<<<(EP + 7) / 8, 256, 0, stream>>>(xl, xr, att + l * D, ei, E, EP,
                                                 e_buf, nmax);
    // 4) exp / denom / weighted scatter-accumulate
    edge_accum<<<(EP + 7) / 8, 256, 0, stream>>>(xl, ei, E, EP, e_buf, nmax,
                                                 denom, acc);
    // 5) normalize + bias + relu + residual
    finalize<<<(tot + 255) / 256, 256, 0, stream>>>(acc, denom, bias + l * D,
                                                    h_in, h_out, n);
    h_in = h_out;
  }
}